// Mesh1_14267881357850
// MI455X (gfx1250) — compile-verified
//
#include <hip/hip_runtime.h>

typedef __attribute__((ext_vector_type(2))) float v2f;
typedef __attribute__((ext_vector_type(8))) float v8f;

#define D_SPATIAL 64
#define D_STRUCT  131
#define D_OUT     256
#define K1  195      // concat(spatial, structural)
#define K1P 196      // padded to multiple of 4
#define K2  131
#define K2P 132

// ---------------------------------------------------------------------------
// Prep: transpose + zero-pad weights into workspace.
//   wt_comb[k][c] = W_comb[c][k]  (k < 195, zero row 195)   [196 x 256]
//   wt_agg [k][c] = W_agg [c][k]  (k < 131, zero row 131)   [132 x 256]
// Makes B-fragment loads coalesced (16 consecutive floats per half-wave) and
// removes all guards from the WMMA K-loop.
// ---------------------------------------------------------------------------
__global__ __launch_bounds__(256)
void prep_weights_kernel(const float* __restrict__ W_comb,
                         const float* __restrict__ W_agg,
                         float* __restrict__ wt_comb,
                         float* __restrict__ wt_agg)
{
    const int i = blockIdx.x * 256 + threadIdx.x;
    if (i < K1P * D_OUT) {
        const int k = i >> 8, c = i & 255;
        wt_comb[i] = (k < K1) ? W_comb[c * K1 + k] : 0.0f;
    }
    if (i < K2P * D_OUT) {
        const int k = i >> 8, c = i & 255;
        wt_agg[i] = (k < K2) ? W_agg[c * K2 + k] : 0.0f;
    }
}

// ---------------------------------------------------------------------------
// Main fused kernel: 1 block = 8 wave32s = 16 rows.
//   out1 = concat(spatial, structural) @ Wt_comb + b_comb   [16 x 256]
//   out2 = mean(self + 3 neighbours)   @ Wt_agg  + b_agg    [16 x 256]
// ---------------------------------------------------------------------------
__global__ __launch_bounds__(256)
void mesh_fused_kernel(const float* __restrict__ spatial,
                       const float* __restrict__ structural,
                       const long long* __restrict__ neighbour,
                       const float* __restrict__ wt_comb,   // [196][256] padded
                       const float* __restrict__ b_comb,
                       const float* __restrict__ wt_agg,    // [132][256] padded
                       const float* __restrict__ b_agg,
                       float* __restrict__ out1,
                       float* __restrict__ out2,
                       int n)
{
    __shared__ __align__(16) float feats[16 * K1P];  // 12.25 KB
    __shared__ __align__(16) float aggf [16 * K2P];  //  8.25 KB

    const int t    = threadIdx.x;          // 0..255
    const int row0 = blockIdx.x * 16;

    // ---- stage concat(spatial, structural) rows into LDS (zero-padded) ----
    for (int i = t; i < 16 * K1P; i += 256) {
        const int r = i / K1P;
        const int k = i - r * K1P;
        int row = row0 + r; if (row >= n) row = n - 1;
        float v = 0.0f;
        if (k < D_SPATIAL)      v = spatial[(long)row * D_SPATIAL + k];
        else if (k < K1)        v = structural[(long)row * D_STRUCT + (k - D_SPATIAL)];
        feats[i] = v;
    }

    // ---- stage neighbour-aggregated features into LDS (zero-padded) ----
    for (int i = t; i < 16 * K2P; i += 256) {
        const int r = i / K2P;
        const int k = i - r * K2P;
        float v = 0.0f;
        if (k < K2) {
            int row = row0 + r; if (row >= n) row = n - 1;
            const long long n0 = neighbour[(long)row * 3 + 0];
            const long long n1 = neighbour[(long)row * 3 + 1];
            const long long n2 = neighbour[(long)row * 3 + 2];
            v = 0.25f * (structural[(long)row * D_STRUCT + k]
                       + structural[n0 * D_STRUCT + k]
                       + structural[n1 * D_STRUCT + k]
                       + structural[n2 * D_STRUCT + k]);
        }
        aggf[i] = v;
    }
    __syncthreads();

    const int lane  = t & 31;
    const int wave  = t >> 5;
    const int mrow  = lane & 15;         // A-row / B-col / C/D-col within tile
    const int koff  = (lane >> 4) * 2;   // K sub-offset (lanes 16-31 hold K=2,3)
    const int mbase = (lane >> 4) * 8;   // C/D row base (lanes 16-31 hold M=8..15)
    const int nc0   = wave * 32;
    const int nc1   = nc0 + 16;
    const bool full = (row0 + 16 <= n);

    // =============== out1: [16 x 196] x [196 x 256] (padded, guard-free) ====
    {
        const float bias0 = b_comb[nc0 + mrow];
        const float bias1 = b_comb[nc1 + mrow];
        v8f c0, c1;
        #pragma unroll
        for (int r = 0; r < 8; ++r) { c0[r] = bias0; c1[r] = bias1; }

        const float* fb = &feats[mrow * K1P + koff];
        const float* w0 = &wt_comb[(long)koff * D_OUT + nc0 + mrow];
        const float* w1 = &wt_comb[(long)koff * D_OUT + nc1 + mrow];

        for (int ki = 0; ki < K1P / 4; ++ki) {
            const int kb = ki * 4;
            const float2 af = *(const float2*)(fb + kb);   // ds_load_b64
            v2f a;  a[0] = af.x;  a[1] = af.y;
            v2f b0, b1;
            b0[0] = w0[(long)kb * D_OUT];
            b0[1] = w0[(long)(kb + 1) * D_OUT];
            b1[0] = w1[(long)kb * D_OUT];
            b1[1] = w1[(long)(kb + 1) * D_OUT];
            c0 = __builtin_amdgcn_wmma_f32_16x16x4_f32(false, a, false, b0, (short)0, c0, false, false);
            c1 = __builtin_amdgcn_wmma_f32_16x16x4_f32(false, a, false, b1, (short)0, c1, false, false);
        }

        float* o0 = out1 + (long)(row0 + mbase) * D_OUT + nc0 + mrow;
        float* o1 = out1 + (long)(row0 + mbase) * D_OUT + nc1 + mrow;
        if (full) {
            #pragma unroll
            for (int r = 0; r < 8; ++r) {
                o0[(long)r * D_OUT] = c0[r];
                o1[(long)r * D_OUT] = c1[r];
            }
        } else {
            #pragma unroll
            for (int r = 0; r < 8; ++r) {
                if (row0 + mbase + r < n) {
                    o0[(long)r * D_OUT] = c0[r];
                    o1[(long)r * D_OUT] = c1[r];
                }
            }
        }
    }

    // =============== out2: [16 x 132] x [132 x 256] (padded, guard-free) ====
    {
        const float bias0 = b_agg[nc0 + mrow];
        const float bias1 = b_agg[nc1 + mrow];
        v8f c0, c1;
        #pragma unroll
        for (int r = 0; r < 8; ++r) { c0[r] = bias0; c1[r] = bias1; }

        const float* fb = &aggf[mrow * K2P + koff];
        const float* w0 = &wt_agg[(long)koff * D_OUT + nc0 + mrow];
        const float* w1 = &wt_agg[(long)koff * D_OUT + nc1 + mrow];

        for (int ki = 0; ki < K2P / 4; ++ki) {
            const int kb = ki * 4;
            const float2 af = *(const float2*)(fb + kb);
            v2f a;  a[0] = af.x;  a[1] = af.y;
            v2f b0, b1;
            b0[0] = w0[(long)kb * D_OUT];
            b0[1] = w0[(long)(kb + 1) * D_OUT];
            b1[0] = w1[(long)kb * D_OUT];
            b1[1] = w1[(long)(kb + 1) * D_OUT];
            c0 = __builtin_amdgcn_wmma_f32_16x16x4_f32(false, a, false, b0, (short)0, c0, false, false);
            c1 = __builtin_amdgcn_wmma_f32_16x16x4_f32(false, a, false, b1, (short)0, c1, false, false);
        }

        float* o0 = out2 + (long)(row0 + mbase) * D_OUT + nc0 + mrow;
        float* o1 = out2 + (long)(row0 + mbase) * D_OUT + nc1 + mrow;
        if (full) {
            #pragma unroll
            for (int r = 0; r < 8; ++r) {
                o0[(long)r * D_OUT] = c0[r];
                o1[(long)r * D_OUT] = c1[r];
            }
        } else {
            #pragma unroll
            for (int r = 0; r < 8; ++r) {
                if (row0 + mbase + r < n) {
                    o0[(long)r * D_OUT] = c0[r];
                    o1[(long)r * D_OUT] = c1[r];
                }
            }
        }
    }
}

extern "C" void kernel_launch(void* const* d_in, const int* in_sizes, int n_in,
                              void* d_out, int out_size, void* d_ws, size_t ws_size,
                              hipStream_t stream) {
    const float*     spatial    = (const float*)d_in[0];
    const float*     structural = (const float*)d_in[1];
    const long long* neighbour  = (const long long*)d_in[2];
    const float*     W_agg      = (const float*)d_in[3];
    const float*     b_agg      = (const float*)d_in[4];
    const float*     W_comb     = (const float*)d_in[5];
    const float*     b_comb     = (const float*)d_in[6];

    const int n = in_sizes[0] / D_SPATIAL;     // N = 200000
    float* out1 = (float*)d_out;               // [N, 256]
    float* out2 = out1 + (long)n * D_OUT;      // [N, 256]

    // workspace layout: wt_comb [196*256] then wt_agg [132*256]  (~336 KB)
    float* wt_comb = (float*)d_ws;
    float* wt_agg  = wt_comb + (long)K1P * D_OUT;

    {
        const int total  = K1P * D_OUT;        // 50176 (covers both)
        const int blocks = (total + 255) / 256;
        prep_weights_kernel<<<blocks, 256, 0, stream>>>(W_comb, W_agg, wt_comb, wt_agg);
    }

    const int blocks = (n + 15) / 16;          // 12500
    mesh_fused_kernel<<<blocks, 256, 0, stream>>>(spatial, structural, neighbour,
                                                  wt_comb, b_comb, wt_agg, b_agg,
                                                  out1, out2, n);
}